// APSDG_47596827574579
// MI455X (gfx1250) — compile-verified
//
#include <hip/hip_runtime.h>
#include <math.h>

#define NN 50000
#define NE 800000
#define DD 128
#define NLAYERS 2

typedef float v2f __attribute__((ext_vector_type(2)));
typedef float v8f __attribute__((ext_vector_type(8)));

__device__ __forceinline__ float wave_sum(float v) {
#pragma unroll
  for (int off = 16; off > 0; off >>= 1)
    v += __shfl_xor(v, off, 32);
  return v;
}

// ---------------- zero ----------------
__global__ void zero_f4_kernel(float4* p, long n4) {
  long i = (long)blockIdx.x * blockDim.x + threadIdx.x;
  long stride = (long)gridDim.x * blockDim.x;
  float4 z = make_float4(0.f, 0.f, 0.f, 0.f);
  for (; i < n4; i += stride) p[i] = z;
}

// ---------------- degree (float, matches maximum(deg,1.0) semantics) ----------------
__global__ void deg_kernel(const int* __restrict__ dst, float* __restrict__ deg, int E) {
  int i = blockIdx.x * blockDim.x + threadIdx.x;
  if (i < E) unsafeAtomicAdd(&deg[dst[i]], 1.0f);
}

// ---------------- per-node pre-maps: e copy, b logmap0, s l2norm ----------------
__global__ void premap_kernel(const float* __restrict__ eIn, const float* __restrict__ bIn,
                              const float* __restrict__ sIn,
                              float* __restrict__ Xe, float* __restrict__ Xb, float* __restrict__ Xs,
                              const float* __restrict__ bCurv) {
  int wave = (blockIdx.x * blockDim.x + threadIdx.x) >> 5;
  int lane = threadIdx.x & 31;
  if (wave >= NN) return;
  long row = (long)wave * DD;
  float sc = sqrtf(bCurv[0]);

  // euclidean: straight copy into GEMM input buffer
  float4 ve = ((const float4*)(eIn + row))[lane];
  ((float4*)(Xe + row))[lane] = ve;

  // hyperbolic: logmap at origin: (2/sc)*atanh(sc*||y||)*y/||y||
  float4 vb = ((const float4*)(bIn + row))[lane];
  float ssb = vb.x * vb.x + vb.y * vb.y + vb.z * vb.z + vb.w * vb.w;
  float an = sqrtf(wave_sum(ssb));
  float anc = fmaxf(an, 1e-20f);
  float t = fminf(sc * anc, 0.9999999f);
  float fb = (2.0f / sc) * atanhf(t) / anc;
  vb.x *= fb; vb.y *= fb; vb.z *= fb; vb.w *= fb;
  ((float4*)(Xb + row))[lane] = vb;

  // spherical: l2 normalize
  float4 vs = ((const float4*)(sIn + row))[lane];
  float sss = vs.x * vs.x + vs.y * vs.y + vs.z * vs.z + vs.w * vs.w;
  float nrm = sqrtf(wave_sum(sss));
  float fs = 1.0f / fmaxf(nrm, 1e-12f);
  vs.x *= fs; vs.y *= fs; vs.z *= fs; vs.w *= fs;
  ((float4*)(Xs + row))[lane] = vs;
}

// ---------------- WMMA GEMM: H = X @ W^T + bias, 3 channels ----------------
// One wave per 16-row strip x ALL 128 columns (8 column tiles). Per k-step (K=4):
// 1 A-fragment load feeds 8 v_wmma_f32_16x16x4_f32 into 8 independent accumulators.
// A (16x4 f32): lanes 0-15 -> M=lane, VGPR0=K0,VGPR1=K1; lanes 16-31 -> K2,K3.
// B (4x16 f32): lanes 0-15 -> N=lane, rows K0/K1; lanes 16-31 -> rows K2/K3.
// B[k][n] = W[n][k], so a float2 from W row n at offset k feeds the B fragment.
__global__ void __launch_bounds__(256) gemm_kernel(
    const float* __restrict__ Xe, const float* __restrict__ Xb, const float* __restrict__ Xs,
    const float* __restrict__ We, const float* __restrict__ Wb, const float* __restrict__ Ws,
    const float* __restrict__ Be, const float* __restrict__ Bb, const float* __restrict__ Bs,
    float* __restrict__ He, float* __restrict__ Hb, float* __restrict__ Hs) {
  const int STRIPS_PER_CH = NN / 16;  // 3125
  int gw = (int)((blockIdx.x * blockDim.x + threadIdx.x) >> 5);
  if (gw >= 3 * STRIPS_PER_CH) return;  // wave-uniform guard (tail waves only)
  int lane = threadIdx.x & 31;

  int ch = gw / STRIPS_PER_CH;
  int mt = gw - ch * STRIPS_PER_CH;  // 0..3124

  const float* X = (ch == 0) ? Xe : ((ch == 1) ? Xb : Xs);
  const float* W = (ch == 0) ? We : ((ch == 1) ? Wb : Ws);
  const float* B = (ch == 0) ? Be : ((ch == 1) ? Bb : Bs);
  float* H = (ch == 0) ? He : ((ch == 1) ? Hb : Hs);

  int nl = lane & 15;
  int kh = lane >> 4;  // 0 or 1: selects K-half of the fragment
  const float* pA = X + (long)(mt * 16 + nl) * DD + kh * 2;
  const float* pB = W + (long)nl * DD + kh * 2;  // column tile nt adds nt*16 rows

  v8f acc[8];
#pragma unroll
  for (int nt = 0; nt < 8; ++nt) acc[nt] = (v8f){};

#pragma unroll 2
  for (int k = 0; k < DD; k += 4) {
    v2f a = *(const v2f*)(pA + k);
    v2f bf[8];
#pragma unroll
    for (int nt = 0; nt < 8; ++nt)
      bf[nt] = *(const v2f*)(pB + (long)nt * 16 * DD + k);
#pragma unroll
    for (int nt = 0; nt < 8; ++nt)
      acc[nt] = __builtin_amdgcn_wmma_f32_16x16x4_f32(
          /*neg_a=*/false, a, /*neg_b=*/false, bf[nt],
          /*c_mod=*/(short)0, acc[nt], /*reuse_a=*/false, /*reuse_b=*/false);
  }

  // C/D layout: VGPR r -> row (mt*16 + r + kh*8), col (nt*16 + nl)
  float* outBase = H + (long)(mt * 16 + kh * 8) * DD + nl;
#pragma unroll
  for (int nt = 0; nt < 8; ++nt) {
    float bias = B[nt * 16 + nl];
    float* o = outBase + nt * 16;
#pragma unroll
    for (int r = 0; r < 8; ++r) o[(long)r * DD] = acc[nt][r] + bias;
  }
}

// ---------------- l2norm rows in place (spherical post-linear) ----------------
__global__ void l2norm_rows_kernel(float* __restrict__ H) {
  int wave = (blockIdx.x * blockDim.x + threadIdx.x) >> 5;
  int lane = threadIdx.x & 31;
  if (wave >= NN) return;
  long row = (long)wave * DD;
  float4 v = ((const float4*)(H + row))[lane];
  float ss = v.x * v.x + v.y * v.y + v.z * v.z + v.w * v.w;
  float nrm = sqrtf(wave_sum(ss));
  float f = 1.0f / fmaxf(nrm, 1e-12f);
  v.x *= f; v.y *= f; v.z *= f; v.w *= f;
  ((float4*)(H + row))[lane] = v;
}

// ---------------- edge scatter-add: one wave per edge, 3 channels ----------------
__global__ void scatter_kernel(const int* __restrict__ src, const int* __restrict__ dst,
                               const float* __restrict__ He, const float* __restrict__ Hb,
                               const float* __restrict__ Hs,
                               float* __restrict__ Se, float* __restrict__ Sb,
                               float* __restrict__ Ss) {
  int wave = (blockIdx.x * blockDim.x + threadIdx.x) >> 5;
  int lane = threadIdx.x & 31;
  if (wave >= NE) return;
  int s = src[wave];
  int d = dst[wave];
  long rs = (long)s * DD;
  long rd = (long)d * DD + lane * 4;

  float4 v;
  v = ((const float4*)(He + rs))[lane];
  unsafeAtomicAdd(Se + rd + 0, v.x);
  unsafeAtomicAdd(Se + rd + 1, v.y);
  unsafeAtomicAdd(Se + rd + 2, v.z);
  unsafeAtomicAdd(Se + rd + 3, v.w);
  v = ((const float4*)(Hb + rs))[lane];
  unsafeAtomicAdd(Sb + rd + 0, v.x);
  unsafeAtomicAdd(Sb + rd + 1, v.y);
  unsafeAtomicAdd(Sb + rd + 2, v.z);
  unsafeAtomicAdd(Sb + rd + 3, v.w);
  v = ((const float4*)(Hs + rs))[lane];
  unsafeAtomicAdd(Ss + rd + 0, v.x);
  unsafeAtomicAdd(Ss + rd + 1, v.y);
  unsafeAtomicAdd(Ss + rd + 2, v.z);
  unsafeAtomicAdd(Ss + rd + 3, v.w);
}

// ---------------- per-node post-maps: mean + leaky / expmap0 / l2norm ----------------
__global__ void postmap_kernel(const float* __restrict__ deg,
                               const float* __restrict__ Se, const float* __restrict__ Sb,
                               const float* __restrict__ Ss,
                               float* __restrict__ oe, float* __restrict__ ob,
                               float* __restrict__ os, int ostride,
                               const float* __restrict__ bCurv) {
  int wave = (blockIdx.x * blockDim.x + threadIdx.x) >> 5;
  int lane = threadIdx.x & 31;
  if (wave >= NN) return;
  long row = (long)wave * DD;
  long orow = (long)wave * ostride;
  float inv = 1.0f / fmaxf(deg[wave], 1.0f);
  float sc = sqrtf(bCurv[0]);

  // euclidean: mean then leaky_relu(0.2)
  float4 v = ((const float4*)(Se + row))[lane];
  v.x *= inv; v.y *= inv; v.z *= inv; v.w *= inv;
  v.x = (v.x > 0.f) ? v.x : 0.2f * v.x;
  v.y = (v.y > 0.f) ? v.y : 0.2f * v.y;
  v.z = (v.z > 0.f) ? v.z : 0.2f * v.z;
  v.w = (v.w > 0.f) ? v.w : 0.2f * v.w;
  ((float4*)(oe + orow))[lane] = v;

  // hyperbolic: mean then expmap0: tanh(sc*||v||/2)/(sc*||v||) * v
  v = ((const float4*)(Sb + row))[lane];
  v.x *= inv; v.y *= inv; v.z *= inv; v.w *= inv;
  float ss = v.x * v.x + v.y * v.y + v.z * v.z + v.w * v.w;
  float vn = sqrtf(wave_sum(ss));
  float vnc = fmaxf(vn, 1e-20f);
  float fac = tanhf(0.5f * sc * vnc) / (sc * vnc);
  v.x *= fac; v.y *= fac; v.z *= fac; v.w *= fac;
  ((float4*)(ob + orow))[lane] = v;

  // spherical: mean then l2norm
  v = ((const float4*)(Ss + row))[lane];
  v.x *= inv; v.y *= inv; v.z *= inv; v.w *= inv;
  ss = v.x * v.x + v.y * v.y + v.z * v.z + v.w * v.w;
  float nrm = sqrtf(wave_sum(ss));
  float f = 1.0f / fmaxf(nrm, 1e-12f);
  v.x *= f; v.y *= f; v.z *= f; v.w *= f;
  ((float4*)(os + orow))[lane] = v;
}

extern "C" void kernel_launch(void* const* d_in, const int* in_sizes, int n_in,
                              void* d_out, int out_size, void* d_ws, size_t ws_size,
                              hipStream_t stream) {
  (void)in_sizes; (void)n_in; (void)out_size; (void)ws_size;
  const int* src = (const int*)d_in[0];
  const int* dst = (const int*)d_in[1];
  const float* e_emb = (const float*)d_in[2];
  const float* b_emb = (const float*)d_in[3];
  const float* s_emb = (const float*)d_in[4];
  const float* e_W = (const float*)d_in[5];
  const float* e_b = (const float*)d_in[6];
  const float* b_W = (const float*)d_in[7];
  const float* b_b = (const float*)d_in[8];
  const float* s_W = (const float*)d_in[9];
  const float* s_b = (const float*)d_in[10];
  const float* b_curv = (const float*)d_in[11];
  // s_curv (d_in[12]) is unused by the reference math (spherical path never uses it)
  float* out = (float*)d_out;

  const long A = (long)NN * DD;  // 6,400,000 floats per [N,128] array
  float* w = (float*)d_ws;
  float* deg = w;                         // 50176 floats (padded)
  float* S = w + 50176;                   // 3*A: Se,Sb,Ss contiguous for one zero pass
  float* Se = S;
  float* Sb = S + A;
  float* Ss = S + 2 * A;
  float* Xe = S + 3 * A;
  float* Xb = Xe + A;
  float* Xs = Xb + A;
  float* He = Xs + A;  // H doubles as "current embeddings" between layers
  float* Hb = He + A;
  float* Hs = Hb + A;

  // degree (computed once; float to match maximum(deg, 1.0))
  zero_f4_kernel<<<64, 256, 0, stream>>>((float4*)deg, 50176 / 4);
  deg_kernel<<<NE / 256, 256, 0, stream>>>(dst, deg, NE);

  const float* curE = e_emb;
  const float* curB = b_emb;
  const float* curS = s_emb;

  const int gemmWaves = 3 * (NN / 16);               // 9375
  const int gemmBlocks = (gemmWaves + 7) / 8;        // 1172

  for (int l = 0; l < NLAYERS; ++l) {
    zero_f4_kernel<<<2048, 256, 0, stream>>>((float4*)S, (3 * A) / 4);
    premap_kernel<<<NN / 8, 256, 0, stream>>>(curE, curB, curS, Xe, Xb, Xs, b_curv);
    gemm_kernel<<<gemmBlocks, 256, 0, stream>>>(
        Xe, Xb, Xs,
        e_W + (long)l * DD * DD, b_W + (long)l * DD * DD, s_W + (long)l * DD * DD,
        e_b + (long)l * DD, b_b + (long)l * DD, s_b + (long)l * DD,
        He, Hb, Hs);
    l2norm_rows_kernel<<<NN / 8, 256, 0, stream>>>(Hs);
    scatter_kernel<<<NE / 8, 256, 0, stream>>>(src, dst, He, Hb, Hs, Se, Sb, Ss);
    if (l < NLAYERS - 1) {
      postmap_kernel<<<NN / 8, 256, 0, stream>>>(deg, Se, Sb, Ss, He, Hb, Hs, DD, b_curv);
      curE = He; curB = Hb; curS = Hs;
    } else {
      postmap_kernel<<<NN / 8, 256, 0, stream>>>(deg, Se, Sb, Ss,
                                                 out, out + DD, out + 2 * DD, 3 * DD, b_curv);
    }
  }
}